// GCN_37520834298162
// MI455X (gfx1250) — compile-verified
//
#include <hip/hip_runtime.h>
#include <math.h>

#define HIDC 128

typedef __attribute__((ext_vector_type(2))) float v2f;
typedef __attribute__((ext_vector_type(8))) float v8f;

// ---------------------------------------------------------------------------
// init: zero layer-1 accumulator, deg = 1.0 (self-loop)
// ---------------------------------------------------------------------------
__global__ void gcn_init(float* __restrict__ acc, float* __restrict__ deg, int N) {
    int idx = blockIdx.x * blockDim.x + threadIdx.x;
    long long total = (long long)N * HIDC;
    if ((long long)idx < total) acc[idx] = 0.0f;
    if (idx < N) deg[idx] = 1.0f;
}

// degree count on dst
__global__ void gcn_degree(const int* __restrict__ dst, float* __restrict__ deg, int E) {
    int e = blockIdx.x * blockDim.x + threadIdx.x;
    if (e < E) atomicAdd(&deg[dst[e]], 1.0f);
}

__global__ void gcn_dinv(const float* __restrict__ deg, float* __restrict__ dinv, int N) {
    int i = blockIdx.x * blockDim.x + threadIdx.x;
    if (i < N) dinv[i] = rsqrtf(deg[i]);
}

// ---------------------------------------------------------------------------
// h = x @ W1 with V_WMMA_F32_16X16X4_F32.
// Block = 256 threads = 8 waves. Block b -> rows [16b, 16b+16).
// Wave w -> cols [16w, 16w+16). K-loop over 128 in steps of 4.
// A tile staged in LDS (stride 132 -> bank-conflict-free).
// Uniform fast path for full tiles (always taken when N % 16 == 0).
// ---------------------------------------------------------------------------
__global__ __launch_bounds__(256) void gcn_gemm1_wmma(const float* __restrict__ x,
                                                      const float* __restrict__ W,
                                                      float* __restrict__ h, int N) {
    __shared__ float As[16 * 132];
    const int t = threadIdx.x;
    const int rowBase = blockIdx.x * 16;           // < 100000, 32-bit safe
    const bool full = (rowBase + 16 <= N);         // scalar, block-uniform

    // cooperative load of 16x128 A tile (512 float4, 2 per thread)
    if (full) {
#pragma unroll
        for (int i = t; i < 16 * 32; i += 256) {
            int r  = i >> 5;
            int c4 = i & 31;
            float4 v = ((const float4*)(x + (size_t)(rowBase + r) * HIDC))[c4];
            float* p = &As[r * 132 + c4 * 4];
            p[0] = v.x; p[1] = v.y; p[2] = v.z; p[3] = v.w;
        }
    } else {
        for (int i = t; i < 16 * 32; i += 256) {
            int r  = i >> 5;
            int c4 = i & 31;
            float4 v = make_float4(0.f, 0.f, 0.f, 0.f);
            if (rowBase + r < N) v = ((const float4*)(x + (size_t)(rowBase + r) * HIDC))[c4];
            float* p = &As[r * 132 + c4 * 4];
            p[0] = v.x; p[1] = v.y; p[2] = v.z; p[3] = v.w;
        }
    }
    __syncthreads();

    const int lane = t & 31;
    const int wv   = t >> 5;           // 0..7 -> 16-col slab
    const int m    = lane & 15;        // A row for this lane
    const int half = lane >> 4;        // 0/1
    const int kl   = half * 2;         // K sub-offset per ISA layout
    const int n    = wv * 16 + (lane & 15);

    v8f c = {};
    const float* Arow = &As[m * 132];
#pragma unroll 8
    for (int k0 = 0; k0 < HIDC; k0 += 4) {
        v2f a, b;
        a.x = Arow[k0 + kl];
        a.y = Arow[k0 + kl + 1];
        b.x = W[(k0 + kl) * HIDC + n];
        b.y = W[(k0 + kl + 1) * HIDC + n];
        c = __builtin_amdgcn_wmma_f32_16x16x4_f32(false, a, false, b,
                                                  (short)0, c, false, false);
    }

    // D layout: VGPR v -> M = v + 8*half, N = lane&15
    float* hp = h + (size_t)(rowBase + 8 * half) * HIDC + n;
    if (full) {
#pragma unroll
        for (int v = 0; v < 8; ++v) hp[v * HIDC] = c[v];
    } else {
#pragma unroll
        for (int v = 0; v < 8; ++v) {
            if (rowBase + 8 * half + v < N) hp[v * HIDC] = c[v];
        }
    }
}

// ---------------------------------------------------------------------------
// Layer-1 message passing: one wave per edge, float4 per lane (32*4 = 128).
// acc[dst] += h[src] * dinv[src]*dinv[dst]   (fp32 global atomics, L2)
// ---------------------------------------------------------------------------
__global__ __launch_bounds__(256) void gcn_agg1(const float* __restrict__ h,
                                                const int* __restrict__ src,
                                                const int* __restrict__ dst,
                                                const float* __restrict__ dinv,
                                                float* __restrict__ acc, int E) {
    int e = blockIdx.x * 8 + (threadIdx.x >> 5);
    if (e >= E) return;
    int lane = threadIdx.x & 31;
    int s = src[e];
    int d = dst[e];
    const float* hs = h + (size_t)s * HIDC;
    __builtin_prefetch(hs + lane * 4, 0, 0);      // global_prefetch_b8
    float nrm = dinv[s] * dinv[d];
    float4 v = ((const float4*)hs)[lane];
    float* o = acc + (size_t)d * HIDC + lane * 4;
    atomicAdd(o + 0, v.x * nrm);
    atomicAdd(o + 1, v.y * nrm);
    atomicAdd(o + 2, v.z * nrm);
    atomicAdd(o + 3, v.w * nrm);
}

// ---------------------------------------------------------------------------
// Finalize layer 1 + layer-2 transform fused (wave per node):
// r = relu(acc + h*dinv^2 + b1);  h2 = dot(r, W2)  via wave32 shuffle reduce.
// ---------------------------------------------------------------------------
__global__ __launch_bounds__(256) void gcn_fin1_dot(const float* __restrict__ acc,
                                                    const float* __restrict__ h,
                                                    const float* __restrict__ dinv,
                                                    const float* __restrict__ b1,
                                                    const float* __restrict__ W2,
                                                    float* __restrict__ h2, int N) {
    int i = blockIdx.x * 8 + (threadIdx.x >> 5);
    if (i >= N) return;
    int lane = threadIdx.x & 31;
    float di = dinv[i];
    float ns = di * di;
    float4 a  = ((const float4*)(acc + (size_t)i * HIDC))[lane];
    float4 hv = ((const float4*)(h   + (size_t)i * HIDC))[lane];
    float4 bb = ((const float4*)b1)[lane];
    float4 w  = ((const float4*)W2)[lane];
    float r0 = fmaxf(a.x + hv.x * ns + bb.x, 0.0f);
    float r1 = fmaxf(a.y + hv.y * ns + bb.y, 0.0f);
    float r2 = fmaxf(a.z + hv.z * ns + bb.z, 0.0f);
    float r3 = fmaxf(a.w + hv.w * ns + bb.w, 0.0f);
    float p = r0 * w.x + r1 * w.y + r2 * w.z + r3 * w.w;
#pragma unroll
    for (int off = 16; off > 0; off >>= 1) p += __shfl_down(p, off, 32);
    if (lane == 0) h2[i] = p;
}

// out[i] = h2[i]*dinv[i]^2 + b2   (self-loop term + bias seeds the accumulator)
__global__ void gcn_init_out(const float* __restrict__ h2, const float* __restrict__ dinv,
                             const float* __restrict__ b2, float* __restrict__ out, int N) {
    int i = blockIdx.x * blockDim.x + threadIdx.x;
    if (i < N) {
        float di = dinv[i];
        out[i] = h2[i] * di * di + b2[0];
    }
}

// Layer-2 message passing: scalar per edge.
__global__ void gcn_agg2(const float* __restrict__ h2, const int* __restrict__ src,
                         const int* __restrict__ dst, const float* __restrict__ dinv,
                         float* __restrict__ out, int E) {
    int e = blockIdx.x * blockDim.x + threadIdx.x;
    if (e < E) {
        int s = src[e];
        int d = dst[e];
        atomicAdd(&out[d], h2[s] * dinv[s] * dinv[d]);
    }
}

// ---------------------------------------------------------------------------
extern "C" void kernel_launch(void* const* d_in, const int* in_sizes, int n_in,
                              void* d_out, int out_size, void* d_ws, size_t ws_size,
                              hipStream_t stream) {
    const float* x  = (const float*)d_in[0];
    const float* W1 = (const float*)d_in[1];
    const float* b1 = (const float*)d_in[2];
    const float* W2 = (const float*)d_in[3];
    const float* b2 = (const float*)d_in[4];
    const int*   ei = (const int*)d_in[5];

    const int N = in_sizes[0] / HIDC;
    const int E = in_sizes[5] / 2;
    const int* src = ei;        // edge_index[0]
    const int* dst = ei + E;    // edge_index[1]
    float* out = (float*)d_out;

    // workspace layout (float4-aligned: big arrays first)
    float* h    = (float*)d_ws;                    // N*128
    float* acc  = h   + (size_t)N * HIDC;          // N*128
    float* deg  = acc + (size_t)N * HIDC;          // N
    float* dinv = deg + N;                         // N
    float* h2   = dinv + N;                        // N

    const int TB = 256;
    long long totNH = (long long)N * HIDC;
    int gInit = (int)((totNH + TB - 1) / TB);

    gcn_init      <<<gInit,             TB, 0, stream>>>(acc, deg, N);
    gcn_degree    <<<(E + TB - 1) / TB, TB, 0, stream>>>(dst, deg, E);
    gcn_dinv      <<<(N + TB - 1) / TB, TB, 0, stream>>>(deg, dinv, N);
    gcn_gemm1_wmma<<<(N + 15) / 16,     TB, 0, stream>>>(x, W1, h, N);
    gcn_agg1      <<<(E + 7) / 8,       TB, 0, stream>>>(h, src, dst, dinv, acc, E);
    gcn_fin1_dot  <<<(N + 7) / 8,       TB, 0, stream>>>(acc, h, dinv, b1, W2, h2, N);
    gcn_init_out  <<<(N + TB - 1) / TB, TB, 0, stream>>>(h2, dinv, b2, out, N);
    gcn_agg2      <<<(E + TB - 1) / TB, TB, 0, stream>>>(h2, src, dst, dinv, out, E);
}